// EligibilityLSTMModule_34978213659000
// MI455X (gfx1250) — compile-verified
//
#include <hip/hip_runtime.h>
#include <math.h>

#define D 4096
#define CLIPV 1.0f

typedef float v2f __attribute__((ext_vector_type(2)));
typedef float v4f __attribute__((ext_vector_type(4)));
typedef float v8f __attribute__((ext_vector_type(8)));

__device__ __forceinline__ float ntload(const float* p) {
    return __builtin_nontemporal_load(p);
}

#define WMMA_ACC(A, B, C) \
    __builtin_amdgcn_wmma_f32_16x16x4_f32(false, (A), false, (B), (short)0, (C), false, false)

// ---------------------------------------------------------------------------
// Phase 1: 9 fused GEMV streams via V_WMMA_F32_16X16X4_F32.
// Each block owns a 16-column tile; its 8 waves split K=4096 into 512-chunks.
// A-matrix (16x4 f32): lane L in [0,15] holds M=L with K=0 (v0), K=1 (v1);
// lanes 16..31 hold K=2 (v0), K=3 (v1). We put x (or r) in row M=0 only, so
// the D-matrix row 0 (acc[0], lanes 0..15) is the GEMV partial for 16 cols.
// B-matrix (4x16 f32): lane(half,col): v0 = W[k+2*half][n], v1 = W[k+2*half+1][n].
// A-operand build is branchless: every lane loads the (cached, 16 KB) x/r
// pair and masks with cndmask -- no per-iteration EXEC save/restore.
// ---------------------------------------------------------------------------
__global__ __launch_bounds__(256) void gates_kernel(
    const float* __restrict__ x, const float* __restrict__ r,
    const float* __restrict__ cell,
    const float* __restrict__ hebb, const float* __restrict__ alpha,
    const float* __restrict__ Wi, const float* __restrict__ bi,
    const float* __restrict__ Wj, const float* __restrict__ bj,
    const float* __restrict__ Wf, const float* __restrict__ bf,
    const float* __restrict__ Wo, const float* __restrict__ bo,
    const float* __restrict__ Whi, const float* __restrict__ bhi,
    const float* __restrict__ Whj, const float* __restrict__ bhj,
    const float* __restrict__ Whf, const float* __restrict__ bhf,
    const float* __restrict__ Who, const float* __restrict__ bho,
    float* __restrict__ out_h, float* __restrict__ out_c,
    float* __restrict__ ws_post)
{
    const int tid  = threadIdx.x;
    const int lane = tid & 31;
    const int wave = tid >> 5;
    const int half = lane >> 4;
    const int col  = lane & 15;
    const int n0   = blockIdx.x * 16;
    const int n    = n0 + col;
    const bool rowA = (col == 0);     // lanes 0 and 16 carry A-matrix row 0

    const float* const Wm[8] = {Wi, Wj, Wf, Wo, Whi, Whj, Whf, Who};

    v8f acc[9];
    const v8f zero8 = {};
#pragma unroll
    for (int s = 0; s < 9; ++s) acc[s] = zero8;

    const int kbeg = wave * (D / 8);
    const int kend = kbeg + (D / 8);
    // B tile base: row (k + 2*half), column n; second VGPR is one row below.
    int idx = (kbeg + 2 * half) * D + n;

    for (int k = kbeg; k < kend; k += 4) {
        // Branchless A build: all lanes load the 8-byte pair (coalesces to two
        // requests per wave; x/r stay L0/L2-resident), then mask.
        const float2 xv = *(const float2*)(x + k + 2 * half);
        const float2 rv = *(const float2*)(r + k + 2 * half);
        v2f ax, ar;
        ax[0] = rowA ? xv.x : 0.f;
        ax[1] = rowA ? xv.y : 0.f;
        ar[0] = rowA ? rv.x : 0.f;
        ar[1] = rowA ? rv.y : 0.f;

        {   // plastic stream: B = alpha (.) hebb ; keep hebb temporal (L2 reuse
            // in phase 3), alpha is single-use -> non-temporal.
            v2f b;
            b[0] = ntload(alpha + idx)     * hebb[idx];
            b[1] = ntload(alpha + idx + D) * hebb[idx + D];
            acc[0] = WMMA_ACC(ax, b, acc[0]);
        }
#pragma unroll
        for (int s = 0; s < 8; ++s) {   // 8 single-use weight streams -> NT
            v2f b;
            b[0] = ntload(Wm[s] + idx);
            b[1] = ntload(Wm[s] + idx + D);
            acc[s + 1] = WMMA_ACC((s < 4) ? ax : ar, b, acc[s + 1]);
        }
        idx += 4 * D;
    }

    // Cross-wave reduction of the 9 row-0 partials through LDS.
    __shared__ float red[9][8][16];
    if (lane < 16) {
#pragma unroll
        for (int s = 0; s < 9; ++s) red[s][wave][lane] = acc[s][0];
    }
    __syncthreads();

    if (tid < 16) {
        float s[9];
#pragma unroll
        for (int ss = 0; ss < 9; ++ss) {
            float t = 0.f;
#pragma unroll
            for (int w = 0; w < 8; ++w) t += red[ss][w][tid];
            s[ss] = t;
        }
        const int nn = n0 + tid;
        const float post = tanhf(s[0] + s[1] + bi[nn] + s[5] + bhi[nn]);
        const float jt   = tanhf(s[2] + bj[nn] + s[6] + bhj[nn]);
        const float ft   = tanhf(s[3] + bf[nn] + s[7] + bhf[nn]);
        const float ot   = tanhf(s[4] + bo[nn] + s[8] + bho[nn]);
        const float c    = ft * cell[nn] + post * jt;
        const float h    = tanhf(c) * ot;
        out_h[nn]   = h;
        out_c[nn]   = c;
        ws_post[nn] = post;
    }
}

// ---------------------------------------------------------------------------
// Phase 2: deterministic single-block reduction  smod = tanh(post.Wmi + bmi)
// ---------------------------------------------------------------------------
__global__ __launch_bounds__(256) void mod_kernel(
    const float* __restrict__ ws_post, const float* __restrict__ Wmi,
    const float* __restrict__ bmi, float* __restrict__ smod)
{
    __shared__ float red[256];
    float t = 0.f;
    for (int i = threadIdx.x; i < D; i += 256) t += ws_post[i] * Wmi[i];
    red[threadIdx.x] = t;
    __syncthreads();
    for (int off = 128; off > 0; off >>= 1) {
        if (threadIdx.x < off) red[threadIdx.x] += red[threadIdx.x + off];
        __syncthreads();
    }
    if (threadIdx.x == 0) *smod = tanhf(red[0] + bmi[0]);
}

// ---------------------------------------------------------------------------
// Phase 3: D x D elementwise trace updates, b128 vectorized.
//   hebb_new = clip(hebb + mod[n]*elig, -1, 1)
//   elig_new = (1-eta)*elig + eta*x[k]*post[n],  mod[n] = smod*Wmo[n]+bmo[n]
// ---------------------------------------------------------------------------
__global__ __launch_bounds__(256) void traces_kernel(
    const float* __restrict__ hebb, const float* __restrict__ elig,
    const float* __restrict__ x, const float* __restrict__ ws_post,
    const float* __restrict__ smod_p,
    const float* __restrict__ Wmo, const float* __restrict__ bmo,
    const float* __restrict__ eta_p,
    float* __restrict__ hebb_out, float* __restrict__ elig_out)
{
    const int i4   = blockIdx.x * 256 + threadIdx.x;  // float4 index
    const int base = i4 * 4;
    const int k    = base >> 12;        // / D
    const int nb   = base & (D - 1);    // % D
    const float smod = *smod_p;
    const float eta  = *eta_p;
    const float omet = 1.0f - eta;
    const float xk   = x[k];

    const v4f wm = *(const v4f*)(Wmo + nb);
    const v4f bm = *(const v4f*)(bmo + nb);
    const v4f ps = *(const v4f*)(ws_post + nb);
    const v4f hb = *(const v4f*)(hebb + base);                       // temporal (L2 hit)
    const v4f el = __builtin_nontemporal_load((const v4f*)(elig + base)); // single use

    v4f mod, hn, en;
#pragma unroll
    for (int c = 0; c < 4; ++c) {
        mod[c] = smod * wm[c] + bm[c];
        float hv = hb[c] + mod[c] * el[c];
        hn[c] = fminf(fmaxf(hv, -CLIPV), CLIPV);
        en[c] = omet * el[c] + eta * xk * ps[c];
    }
    __builtin_nontemporal_store(hn, (v4f*)(hebb_out + base));
    __builtin_nontemporal_store(en, (v4f*)(elig_out + base));
}

extern "C" void kernel_launch(void* const* d_in, const int* in_sizes, int n_in,
                              void* d_out, int out_size, void* d_ws, size_t ws_size,
                              hipStream_t stream) {
    const float* x    = (const float*)d_in[0];
    const float* r    = (const float*)d_in[1];
    const float* cell = (const float*)d_in[2];
    const float* hebb = (const float*)d_in[3];
    const float* elig = (const float*)d_in[4];
    const float* Wi  = (const float*)d_in[5];  const float* bi  = (const float*)d_in[6];
    const float* Wj  = (const float*)d_in[7];  const float* bj  = (const float*)d_in[8];
    const float* Wf  = (const float*)d_in[9];  const float* bf  = (const float*)d_in[10];
    const float* Wo  = (const float*)d_in[11]; const float* bo  = (const float*)d_in[12];
    const float* Whi = (const float*)d_in[13]; const float* bhi = (const float*)d_in[14];
    const float* Whj = (const float*)d_in[15]; const float* bhj = (const float*)d_in[16];
    const float* Whf = (const float*)d_in[17]; const float* bhf = (const float*)d_in[18];
    const float* Who = (const float*)d_in[19]; const float* bho = (const float*)d_in[20];
    const float* Wmi = (const float*)d_in[21]; const float* bmi = (const float*)d_in[22];
    const float* Wmo = (const float*)d_in[23]; const float* bmo = (const float*)d_in[24];
    const float* eta = (const float*)d_in[25];
    const float* alpha = (const float*)d_in[26];

    float* out      = (float*)d_out;
    float* out_h    = out;                                  // [D]
    float* hebb_out = out + D;                              // [D*D]
    float* elig_out = out + D + (size_t)D * D;              // [D*D]
    float* out_c    = out + D + 2 * (size_t)D * D;          // [D]

    float* ws      = (float*)d_ws;
    float* ws_post = ws;        // D floats
    float* smod    = ws + D;    // 1 float

    gates_kernel<<<D / 16, 256, 0, stream>>>(
        x, r, cell, hebb, alpha,
        Wi, bi, Wj, bj, Wf, bf, Wo, bo,
        Whi, bhi, Whj, bhj, Whf, bhf, Who, bho,
        out_h, out_c, ws_post);

    mod_kernel<<<1, 256, 0, stream>>>(ws_post, Wmi, bmi, smod);

    traces_kernel<<<((size_t)D * D) / (256 * 4), 256, 0, stream>>>(
        hebb, elig, x, ws_post, smod, Wmo, bmo, eta, hebb_out, elig_out);
}